// GHCF_11905649344756
// MI455X (gfx1250) — compile-verified
//
#include <hip/hip_runtime.h>
#include <hip/hip_bf16.h>

// ---------------------------------------------------------------------------
// GHCF on MI455X (gfx1250): scatter/gather graph conv + fp32 WMMA GEMMs.
// D = 64 everywhere. W_PV == 0 -> pv graph convolutions are exact zeros and
// are skipped entirely (600k edges saved); pv only contributes via the tiny
// edge-vector chain + broadcast output.
// ---------------------------------------------------------------------------

typedef float v2f __attribute__((ext_vector_type(2)));
typedef float v8f __attribute__((ext_vector_type(8)));

#define NUM_USER 200000
#define NUM_ITEM 100000
#define E_BUY    200000
#define E_CART   400000
#define E_PV     600000
#define DD       64

__device__ __forceinline__ void atomAddF(float* p, float v) {
  // lowers to global_atomic_add_f32 on gfx1250 (no CAS loop)
  unsafeAtomicAdd(p, v);
}

// ---------------- generic helpers ----------------

__global__ void zero_kernel(float* __restrict__ p, size_t n) {
  size_t i = (size_t)blockIdx.x * blockDim.x + threadIdx.x;
  size_t st = (size_t)gridDim.x * blockDim.x;
  for (; i < n; i += st) p[i] = 0.0f;
}

__global__ void scale_inplace_kernel(float* __restrict__ p, size_t n, float s) {
  size_t i = (size_t)blockIdx.x * blockDim.x + threadIdx.x;
  size_t st = (size_t)gridDim.x * blockDim.x;
  for (; i < n; i += st) p[i] *= s;
}

// v = in*0.01 written to both the working buffer and the accumulator.
__global__ void init_scaled_kernel(const float* __restrict__ in,
                                   float* __restrict__ cur,
                                   float* __restrict__ acc, size_t n) {
  size_t i = (size_t)blockIdx.x * blockDim.x + threadIdx.x;
  size_t st = (size_t)gridDim.x * blockDim.x;
  for (; i < n; i += st) {
    float v = in[i] * 0.01f;
    cur[i] = v;
    acc[i] = v;
  }
}

// out[row][c] = vec[c] * 0.2   (row-broadcast of the edge-type vector)
__global__ void broadcast_rows_kernel(const float* __restrict__ vec,
                                      float* __restrict__ out, size_t total) {
  size_t i = (size_t)blockIdx.x * blockDim.x + threadIdx.x;
  size_t st = (size_t)gridDim.x * blockDim.x;
  for (; i < total; i += st) out[i] = vec[i & 63] * 0.2f;
}

// ---------------- degrees ----------------

__global__ void count_degrees_kernel(const int* __restrict__ src,
                                     const int* __restrict__ dst,
                                     float* __restrict__ degU,
                                     float* __restrict__ degI, int nE) {
  int e = blockIdx.x * blockDim.x + threadIdx.x;
  if (e < nE) {
    atomAddF(&degU[src[e]], 1.0f);
    atomAddF(&degI[dst[e]], 1.0f);
  }
}

__global__ void rsqrt_deg_kernel(float* __restrict__ p, size_t n) {
  size_t i = (size_t)blockIdx.x * blockDim.x + threadIdx.x;
  size_t st = (size_t)gridDim.x * blockDim.x;
  for (; i < n; i += st) p[i] = rsqrtf(fmaxf(p[i], 1.0f));
}

// ---------------- edge-type vector chain (tiny, one block) ----------------
// ef = edges_emb[e_type]*0.01 ; eb=ef[0], ep=ef[1], ec=ef[2]
// per layer: e <- e @ EW_k ; acc += e.  Stores stage-k vectors (k=0..3) for
// use inside the graph conv layers, and the 5-term accumulated vectors.
__global__ void edge_chain_kernel(const float* __restrict__ edges_emb,
                                  const int* __restrict__ e_type,
                                  const float* __restrict__ ew0,
                                  const float* __restrict__ ew1,
                                  const float* __restrict__ ew2,
                                  const float* __restrict__ ew3,
                                  float* __restrict__ eVecs,  // [3][4][64]
                                  float* __restrict__ eAcc) { // [3][64]
  __shared__ float cur[3][64];
  __shared__ float nxt[3][64];
  __shared__ float acc[3][64];
  int n = threadIdx.x;
  if (n < 64) {
    for (int t = 0; t < 3; ++t) {
      float v = edges_emb[e_type[t] * 64 + n] * 0.01f;
      cur[t][n] = v;
      acc[t][n] = v;
      eVecs[(t * 4 + 0) * 64 + n] = v;
    }
  }
  __syncthreads();
  const float* ews[4] = {ew0, ew1, ew2, ew3};
  for (int k = 0; k < 4; ++k) {
    if (n < 64) {
      const float* EW = ews[k];
      for (int t = 0; t < 3; ++t) {
        float sum = 0.0f;
        for (int j = 0; j < 64; ++j) sum += cur[t][j] * EW[j * 64 + n];
        nxt[t][n] = sum;
      }
    }
    __syncthreads();
    if (n < 64) {
      for (int t = 0; t < 3; ++t) {
        cur[t][n] = nxt[t][n];
        acc[t][n] += nxt[t][n];
        if (k < 3) eVecs[(t * 4 + (k + 1)) * 64 + n] = nxt[t][n];
      }
    }
    __syncthreads();
  }
  if (n < 64)
    for (int t = 0; t < 3; ++t) eAcc[t * 64 + n] = acc[t][n];
}

// ---------------- edge scatter (both directions in one pass) ----------------
// One wave per edge; lane handles 2 columns (float2 => 256B per wave per row).
//   fwd: aggI[dst] += s[src] * rsU[src] * eVec       (user -> item)
//   rev: aggU[src] += d[dst] * rsI[dst] * eVec       (item -> user)
__launch_bounds__(256)
__global__ void scatter_edges_kernel(const int* __restrict__ src,
                                     const int* __restrict__ dst,
                                     const float* __restrict__ s,
                                     const float* __restrict__ d,
                                     const float* __restrict__ rsU,
                                     const float* __restrict__ rsI,
                                     const float* __restrict__ eVec,
                                     float* __restrict__ aggI,
                                     float* __restrict__ aggU, int nE) {
  const int wpb = blockDim.x >> 5;
  const int lane = threadIdx.x & 31;
  const int c0 = lane * 2;
  int wid = blockIdx.x * wpb + (threadIdx.x >> 5);
  const int totalWaves = gridDim.x * wpb;
  const float e0 = eVec[c0];
  const float e1 = eVec[c0 + 1];
  for (int e = wid; e < nE; e += totalWaves) {
    const int u = src[e];
    const int i = dst[e];
    const float ru = rsU[u];
    const float ri = rsI[i];
    const v2f sv = *(const v2f*)(s + (size_t)u * DD + c0);
    const v2f dv = *(const v2f*)(d + (size_t)i * DD + c0);
    atomAddF(&aggI[(size_t)i * DD + c0 + 0], sv.x * ru * e0);
    atomAddF(&aggI[(size_t)i * DD + c0 + 1], sv.y * ru * e1);
    atomAddF(&aggU[(size_t)u * DD + c0 + 0], dv.x * ri * e0);
    atomAddF(&aggU[(size_t)u * DD + c0 + 1], dv.y * ri * e1);
  }
}

// ---------------- fused GEMM + LeakyReLU + weighted combine ----------------
// out[r,:] = wA*lrelu((aggA[r,:]*rsA[r]) @ W) + wB*lrelu((aggB[r,:]*rsB[r]) @ W)
// written to the next-layer embedding and added into the running accumulator.
// One wave per 16-row tile, V_WMMA_F32_16X16X4_F32, K=64 in 16 steps,
// N=64 in 4 tiles.  M is a multiple of 16 (100000 and 200000 both are).
__launch_bounds__(256)
__global__ void gemm_lrelu_combine_kernel(const float* __restrict__ aggA,
                                          const float* __restrict__ aggB,
                                          const float* __restrict__ rsA,
                                          const float* __restrict__ rsB,
                                          const float* __restrict__ W,
                                          float wA, float wB,
                                          float* __restrict__ outCur,
                                          float* __restrict__ outAcc, int M) {
  __shared__ float Ws[DD * DD];
  for (int i = threadIdx.x; i < DD * DD; i += blockDim.x) Ws[i] = W[i];
  __syncthreads();

  const int wave = threadIdx.x >> 5;
  const int lane = threadIdx.x & 31;
  const int hi = lane >> 4;  // lanes 16..31 hold K+2 / M+8 halves
  const int m = lane & 15;
  const int nTiles = M >> 4;
  const int tile = blockIdx.x * (blockDim.x >> 5) + wave;
  if (tile >= nTiles) return;  // wave-uniform: EXEC all-1s inside WMMA
  const int row0 = tile << 4;
  const int row = row0 + m;

  const v8f vzero = {0.f, 0.f, 0.f, 0.f, 0.f, 0.f, 0.f, 0.f};
  v8f out[4];
#pragma unroll
  for (int nt = 0; nt < 4; ++nt) out[nt] = vzero;

  const float* aggs[2] = {aggA, aggB};
  const float* rss[2] = {rsA, rsB};
  const float wts[2] = {wA, wB};

  for (int t = 0; t < 2; ++t) {
    const float rs = rss[t][row];
    const float* arow = aggs[t] + (size_t)row * DD;
    // A fragments: lane(hi,m) holds A[m][ks*4 + hi*2 + {0,1}] (ISA 16x4 layout)
    v2f af[16];
#pragma unroll
    for (int ks = 0; ks < 16; ++ks) {
      v2f a = *(const v2f*)(arow + ks * 4 + hi * 2);
      af[ks] = a * rs;
    }
#pragma unroll
    for (int nt = 0; nt < 4; ++nt) {
      v8f c = vzero;
#pragma unroll
      for (int ks = 0; ks < 16; ++ks) {
        const int k0 = ks * 4 + hi * 2;
        v2f b;
        b.x = Ws[(k0 + 0) * DD + nt * 16 + m];
        b.y = Ws[(k0 + 1) * DD + nt * 16 + m];
        c = __builtin_amdgcn_wmma_f32_16x16x4_f32(
            false, af[ks], false, b, (short)0, c, false, false);
      }
      const float wt = wts[t];
#pragma unroll
      for (int v = 0; v < 8; ++v) {
        float x = c[v];
        x = x > 0.0f ? x : 0.01f * x;   // LeakyReLU(0.01)
        out[nt][v] += wt * x;
      }
    }
  }

  // C/D layout: lane(hi,m) vgpr v -> element (row0 + v + 8*hi, nt*16 + m)
#pragma unroll
  for (int nt = 0; nt < 4; ++nt) {
    const int col = nt * 16 + m;
#pragma unroll
    for (int v = 0; v < 8; ++v) {
      const int r = row0 + v + hi * 8;
      const size_t idx = (size_t)r * DD + col;
      const float val = out[nt][v];
      outCur[idx] = val;
      outAcc[idx] += val;
    }
  }
}

// ---------------------------------------------------------------------------

extern "C" void kernel_launch(void* const* d_in, const int* in_sizes, int n_in,
                              void* d_out, int out_size, void* d_ws,
                              size_t ws_size, hipStream_t stream) {
  (void)in_sizes; (void)n_in; (void)out_size; (void)ws_size;

  const float* user_emb  = (const float*)d_in[0];
  const float* item_emb  = (const float*)d_in[1];
  const float* edges_emb = (const float*)d_in[2];
  const float* Wk[4]  = {(const float*)d_in[3], (const float*)d_in[4],
                         (const float*)d_in[5], (const float*)d_in[6]};
  const float* EWk[4] = {(const float*)d_in[7], (const float*)d_in[8],
                         (const float*)d_in[9], (const float*)d_in[10]};
  const int* buy_src  = (const int*)d_in[11];
  const int* buy_dst  = (const int*)d_in[12];
  const int* cart_src = (const int*)d_in[13];
  const int* cart_dst = (const int*)d_in[14];
  const int* e_type   = (const int*)d_in[17];

  const size_t SZ_S = (size_t)NUM_USER * DD;  // 12.8M floats
  const size_t SZ_D = (size_t)NUM_ITEM * DD;  //  6.4M floats

  // ---- workspace layout (~231 MB) ----
  float* ws = (float*)d_ws;
  float* s        = ws;                       // current user embedding
  float* dbuf     = s + SZ_S;                 // current item embedding
  float* rsU_buy  = dbuf + SZ_D;              // degrees -> rsqrt in place
  float* rsI_buy  = rsU_buy + NUM_USER;
  float* rsU_cart = rsI_buy + NUM_ITEM;
  float* rsI_cart = rsU_cart + NUM_USER;
  float* aggU_buy  = rsI_cart + NUM_ITEM;     // 4 agg buffers, contiguous
  float* aggU_cart = aggU_buy + SZ_S;
  float* aggI_buy  = aggU_cart + SZ_S;
  float* aggI_cart = aggI_buy + SZ_D;
  float* eVecs = aggI_cart + SZ_D;            // [3][4][64]
  float* eAcc  = eVecs + 3 * 4 * 64;          // [3][64]

  // ---- output layout (return-order concat) ----
  float* out      = (float*)d_out;
  float* sAcc     = out;                      // s_acc (unscaled until end)
  float* dAcc     = out + SZ_S;               // d_acc
  float* eBuyOut  = dAcc + SZ_D;              // E_BUY x 64
  float* eCartOut = eBuyOut + (size_t)E_BUY * DD;
  float* ePvOut   = eCartOut + (size_t)E_CART * DD;

  const size_t DEG_TOTAL = 2 * (size_t)NUM_USER + 2 * (size_t)NUM_ITEM;

  // 1) degrees (graph is fixed across layers -> once)
  zero_kernel<<<2048, 256, 0, stream>>>(rsU_buy, DEG_TOTAL);
  count_degrees_kernel<<<(E_BUY + 255) / 256, 256, 0, stream>>>(
      buy_src, buy_dst, rsU_buy, rsI_buy, E_BUY);
  count_degrees_kernel<<<(E_CART + 255) / 256, 256, 0, stream>>>(
      cart_src, cart_dst, rsU_cart, rsI_cart, E_CART);
  rsqrt_deg_kernel<<<2048, 256, 0, stream>>>(rsU_buy, DEG_TOTAL);

  // 2) initial embeddings (also seed accumulators in d_out)
  init_scaled_kernel<<<4096, 256, 0, stream>>>(user_emb, s, sAcc, SZ_S);
  init_scaled_kernel<<<4096, 256, 0, stream>>>(item_emb, dbuf, dAcc, SZ_D);

  // 3) edge-type vector chains (tiny)
  edge_chain_kernel<<<1, 64, 0, stream>>>(edges_emb, e_type, EWk[0], EWk[1],
                                          EWk[2], EWk[3], eVecs, eAcc);

  const float W_BUY = 1.0f / 6.0f;
  const float W_CART = 5.0f / 6.0f;

  // 4) four propagation layers (pv skipped: weight is exactly 0)
  for (int k = 0; k < 4; ++k) {
    zero_kernel<<<8192, 256, 0, stream>>>(aggU_buy, 2 * SZ_S + 2 * SZ_D);

    scatter_edges_kernel<<<8192, 256, 0, stream>>>(
        buy_src, buy_dst, s, dbuf, rsU_buy, rsI_buy,
        eVecs + (0 * 4 + k) * 64, aggI_buy, aggU_buy, E_BUY);
    scatter_edges_kernel<<<8192, 256, 0, stream>>>(
        cart_src, cart_dst, s, dbuf, rsU_cart, rsI_cart,
        eVecs + (2 * 4 + k) * 64, aggI_cart, aggU_cart, E_CART);

    // item side: d = 1/6*lrelu(buy@W) + 5/6*lrelu(cart@W); dAcc += d
    gemm_lrelu_combine_kernel<<<(NUM_ITEM / 16 + 7) / 8, 256, 0, stream>>>(
        aggI_buy, aggI_cart, rsI_buy, rsI_cart, Wk[k], W_BUY, W_CART,
        dbuf, dAcc, NUM_ITEM);
    // user side
    gemm_lrelu_combine_kernel<<<(NUM_USER / 16 + 7) / 8, 256, 0, stream>>>(
        aggU_buy, aggU_cart, rsU_buy, rsU_cart, Wk[k], W_BUY, W_CART,
        s, sAcc, NUM_USER);
  }

  // 5) s_acc/5, d_acc/5
  scale_inplace_kernel<<<4096, 256, 0, stream>>>(out, SZ_S + SZ_D, 0.2f);

  // 6) broadcast edge-type outputs (buy=type0, pv=type1, cart=type2)
  broadcast_rows_kernel<<<4096, 256, 0, stream>>>(eAcc + 0 * 64, eBuyOut,
                                                  (size_t)E_BUY * DD);
  broadcast_rows_kernel<<<4096, 256, 0, stream>>>(eAcc + 2 * 64, eCartOut,
                                                  (size_t)E_CART * DD);
  broadcast_rows_kernel<<<4096, 256, 0, stream>>>(eAcc + 1 * 64, ePvOut,
                                                  (size_t)E_PV * DD);
}